// StructuralSimilarityLoss_38654705664142
// MI455X (gfx1250) — compile-verified
//
#include <hip/hip_runtime.h>
#include <hip/hip_bf16.h>

typedef __attribute__((ext_vector_type(2))) float v2f;
typedef __attribute__((ext_vector_type(8))) float v8f;

#define IMG_H   512
#define IMG_W   512
#define NBATCH  32
#define TILE_H  32
#define TILE_W  64
#define HALO    5
#define KSZ     11

#define FH      42            // valid staged rows (TILE_H + 2*HALO)
#define FW      74            // valid staged cols (TILE_W + 2*HALO)
#define FH_AL   48            // padded to 3 x 16 row-tiles (rows 42..47 zeroed)
#define FS      76            // F row stride (floats): conflict-free b64 A loads
#define FSZ     (FH_AL*FS)    // 3648
#define HH      48            // horizontal-result rows (3 x 16)
#define HS      72            // H row stride: (8k+n)%64 -> conflict-free B loads
#define HSZ     (HH*HS)       // 3456
#define VS      64            // vertical-result stride == TILE_W (dense)
#define VSZ     (TILE_H*VS)   // 2048

#define GOFF    0             // 1D gaussian (11 floats)
#define ROFF    16            // per-wave reduction slots (8)
#define FOFF    32
#define HOFF    (FOFF + 5*FSZ)          // 18272
#define SMEM_FLOATS (HOFF + 5*HSZ)      // 35552 floats = 142208 bytes

#define C1F     1.0e-4f
#define C2F     9.0e-4f

// Per-wave 16x16 output tile of a SAME 11-tap conv expressed as a banded matmul
// Out(16x16) = In_ext(16x28) x Band(28x16), split into 7 chained K=4 f32 WMMAs.

__global__ __launch_bounds__(256) void ssim_tile_kernel(
    const float* __restrict__ pred, const float* __restrict__ tgt,
    const float* __restrict__ win, float* __restrict__ partial)
{
    extern __shared__ float smem[];
    const int tid  = threadIdx.x;
    const int lane = tid & 31;
    const int wid  = tid >> 5;
    const int bx0  = blockIdx.x * TILE_W;
    const int by0  = blockIdx.y * TILE_H;
    const int b    = blockIdx.z;

    // ---- 1D gaussian taps: g[i] = sum_j window[i][j]  (sum(g) == 1) ----
    if (tid < KSZ) {
        float s = 0.f;
        #pragma unroll
        for (int j = 0; j < KSZ; ++j) s += win[tid * KSZ + j];
        smem[GOFF + tid] = s;
    }
    __syncthreads();

    // ---- stage 5 fields (p, t, p^2, t^2, p*t) over padded halo region ----
    // Branchless: clamped addresses (always valid) + zero-select for pad/outside.
    const float* p0 = pred + (size_t)b * 3 * IMG_H * IMG_W;
    const float* t0 = tgt  + (size_t)b * 3 * IMG_H * IMG_W;
    for (int idx = tid; idx < FH_AL * FS; idx += 256) {
        int r = idx / FS, c = idx - r * FS;
        int gy = by0 - HALO + r, gx = bx0 - HALO + c;
        bool ok = (r < FH) & (c < FW) &
                  ((unsigned)gy < IMG_H) & ((unsigned)gx < IMG_W);
        int oy = min(max(gy, 0), IMG_H - 1);
        int ox = min(max(gx, 0), IMG_W - 1);
        int o  = oy * IMG_W + ox;
        float p = (p0[o] + p0[o + IMG_H*IMG_W] + p0[o + 2*IMG_H*IMG_W]) * (1.f/3.f);
        float t = (t0[o] + t0[o + IMG_H*IMG_W] + t0[o + 2*IMG_H*IMG_W]) * (1.f/3.f);
        p = ok ? p : 0.f;
        t = ok ? t : 0.f;
        smem[FOFF + 0*FSZ + idx] = p;
        smem[FOFF + 1*FSZ + idx] = t;
        smem[FOFF + 2*FSZ + idx] = p * p;
        smem[FOFF + 3*FSZ + idx] = t * t;
        smem[FOFF + 4*FSZ + idx] = p * t;
    }

    // ---- per-lane banded weights: value g[k-x], x = lane&15, k = 4*kc + koff(+1) ----
    // Identical per-lane packing serves as B (horizontal) and A (vertical):
    //   A 16x4 f32: lanes0-15 M=0..15, v0:K=0 v1:K=1; lanes16-31 v0:K=2 v1:K=3
    //   B 4x16 f32: lanes0-15 N=0..15, v0:K=0 v1:K=1; lanes16-31 v0:K=2 v1:K=3
    const int x    = lane & 15;
    const int koff = (lane >> 4) << 1;   // 0 or 2
    v2f wc[7];
    #pragma unroll
    for (int kc = 0; kc < 7; ++kc) {
        int k0 = kc * 4 + koff;
        int d0 = k0 - x, d1 = k0 + 1 - x;
        wc[kc][0] = (d0 >= 0 && d0 < KSZ) ? smem[GOFF + d0] : 0.f;
        wc[kc][1] = (d1 >= 0 && d1 < KSZ) ? smem[GOFF + d1] : 0.f;
    }
    __syncthreads();

    // ---- horizontal conv: 5 fields x 3 row-tiles x 4 col-tiles = 60 wave-tasks ----
    const int cst = (lane >> 4) << 3;    // D rows: +8 for upper lane half
    for (int t = wid; t < 60; t += 8) {
        int f = t / 12, rem = t % 12, i = rem >> 2, j = rem & 3;
        int abase = FOFF + f*FSZ + (16*i + x)*FS + 16*j + koff;   // even -> 8B aligned
        v8f acc = {};
        #pragma unroll
        for (int kc = 0; kc < 7; ++kc) {
            v2f a = *reinterpret_cast<const v2f*>(&smem[abase + kc*4]);  // ds_load_b64
            acc = __builtin_amdgcn_wmma_f32_16x16x4_f32(
                false, a, false, wc[kc], (short)0, acc, false, false);
        }
        int hbase = HOFF + f*HSZ + (16*i + cst)*HS + 16*j + x;
        #pragma unroll
        for (int r = 0; r < 8; ++r) smem[hbase + r*HS] = acc[r];
    }
    __syncthreads();

    // ---- vertical conv: 5 fields x 2 row-tiles x 4 col-tiles = 40 wave-tasks ----
    // results overwrite the (now dead) F region, stride 64 (dense 32x64 per field)
    for (int t = wid; t < 40; t += 8) {
        int f = t / 8, rem = t % 8, i = rem >> 2, j = rem & 3;
        int bbase = HOFF + f*HSZ + (16*i + koff)*HS + 16*j + x;
        v8f acc = {};
        #pragma unroll
        for (int kc = 0; kc < 7; ++kc) {
            v2f bm;
            bm[0] = smem[bbase + kc*4*HS];
            bm[1] = smem[bbase + kc*4*HS + HS];
            acc = __builtin_amdgcn_wmma_f32_16x16x4_f32(
                false, wc[kc], false, bm, (short)0, acc, false, false);
        }
        int vbase = FOFF + f*VSZ + (16*i + cst)*VS + 16*j + x;
        #pragma unroll
        for (int r = 0; r < 8; ++r) smem[vbase + r*VS] = acc[r];
    }
    __syncthreads();

    // ---- SSIM map + block reduction ----
    float lsum = 0.f;
    for (int idx = tid; idx < TILE_H * TILE_W; idx += 256) {
        float mu1 = smem[FOFF + 0*VSZ + idx];
        float mu2 = smem[FOFF + 1*VSZ + idx];
        float ep2 = smem[FOFF + 2*VSZ + idx];
        float et2 = smem[FOFF + 3*VSZ + idx];
        float ept = smem[FOFF + 4*VSZ + idx];
        float m11 = mu1 * mu1, m22 = mu2 * mu2, m12 = mu1 * mu2;
        float s1 = ep2 - m11, s2 = et2 - m22, s12 = ept - m12;
        float num = (2.f * m12 + C1F) * (2.f * s12 + C2F);
        float den = (m11 + m22 + C1F) * (s1 + s2 + C2F);
        lsum += 1.f - num / den;
    }
    #pragma unroll
    for (int off = 16; off > 0; off >>= 1) lsum += __shfl_down(lsum, off, 32);
    if (lane == 0) smem[ROFF + wid] = lsum;
    __syncthreads();
    if (tid == 0) {
        float s = 0.f;
        #pragma unroll
        for (int w = 0; w < 8; ++w) s += smem[ROFF + w];
        partial[(blockIdx.z * gridDim.y + blockIdx.y) * gridDim.x + blockIdx.x] = s;
    }
}

__global__ __launch_bounds__(256) void ssim_reduce_kernel(
    const float* __restrict__ partial, float* __restrict__ out, int n, float scale)
{
    __shared__ float red[8];
    float s = 0.f;
    for (int i = threadIdx.x; i < n; i += 256) s += partial[i];
    #pragma unroll
    for (int off = 16; off > 0; off >>= 1) s += __shfl_down(s, off, 32);
    int lane = threadIdx.x & 31, wid = threadIdx.x >> 5;
    if (lane == 0) red[wid] = s;
    __syncthreads();
    if (threadIdx.x == 0) {
        float tot = 0.f;
        #pragma unroll
        for (int w = 0; w < 8; ++w) tot += red[w];
        out[0] = tot * scale;
    }
}

extern "C" void kernel_launch(void* const* d_in, const int* in_sizes, int n_in,
                              void* d_out, int out_size, void* d_ws, size_t ws_size,
                              hipStream_t stream) {
    (void)in_sizes; (void)n_in; (void)out_size; (void)ws_size;
    const float* pred = (const float*)d_in[0];
    const float* tgt  = (const float*)d_in[1];
    const float* win  = (const float*)d_in[2];
    float* partial = (float*)d_ws;

    const size_t shmem = SMEM_FLOATS * sizeof(float);   // 142,208 B (2 WGs / 320 KB WGP)
    (void)hipFuncSetAttribute(reinterpret_cast<const void*>(ssim_tile_kernel),
                              hipFuncAttributeMaxDynamicSharedMemorySize, (int)shmem);

    dim3 grid(IMG_W / TILE_W, IMG_H / TILE_H, NBATCH);  // (8, 16, 32) = 4096 blocks
    ssim_tile_kernel<<<grid, 256, shmem, stream>>>(pred, tgt, win, partial);

    const int nblocks = (IMG_W / TILE_W) * (IMG_H / TILE_H) * NBATCH;
    const float scale = 1.f / ((float)NBATCH * IMG_H * IMG_W);
    ssim_reduce_kernel<<<1, 256, 0, stream>>>(partial, (float*)d_out, nblocks, scale);
}